// GCNModel_91053306675808
// MI455X (gfx1250) — compile-verified
//
#include <hip/hip_runtime.h>
#include <hip/hip_bf16.h>

typedef __attribute__((ext_vector_type(2))) float v2f;
typedef __attribute__((ext_vector_type(8))) float v8f;

__device__ __forceinline__ v8f wmma4(v2f a, v2f b, v8f c) {
  // D = A(16x4) * B(4x16) + C(16x16), full f32
  return __builtin_amdgcn_wmma_f32_16x16x4_f32(false, a, false, b, (short)0, c,
                                               false, false);
}

__device__ __forceinline__ v8f vzero8() {
  v8f z = {0.f, 0.f, 0.f, 0.f, 0.f, 0.f, 0.f, 0.f};
  return z;
}

// Hardware FP32 atomic add (global_atomic_add_f32, RNE in L2) — avoids the
// CAS-loop fallback of plain atomicAdd(float*).
__device__ __forceinline__ void fadd_atomic(float* p, float v) {
  unsafeAtomicAdd(p, v);
}

// ---------------- degree / normalization ----------------
__global__ void k_deg_init(float* __restrict__ deg, int n) {
  int i = blockIdx.x * blockDim.x + threadIdx.x;
  if (i < n) deg[i] = 1.0f;  // self-loop
}

__global__ void k_deg_edges(const int* __restrict__ dst, float* __restrict__ deg,
                            int e) {
  int i = blockIdx.x * blockDim.x + threadIdx.x;
  if (i < e) fadd_atomic(&deg[dst[i]], 1.0f);
}

__global__ void k_deg_rsqrt(float* __restrict__ deg, int n) {
  int i = blockIdx.x * blockDim.x + threadIdx.x;
  if (i < n) deg[i] = rsqrtf(deg[i]);  // deg >= 1 always
}

// ---------------- dense GEMM: C[M,NC] = (relu?)(A[M,K]) @ B[K,NC] ----------------
// One wave computes one 16x16 C tile with V_WMMA_F32_16X16X4_F32.
// A-layout: lane m = lane&15 holds row m; half = lane>>4 selects K pair.
// C/D layout: VGPR r -> row r (lane<16) / r+8 (lane>=16), col = lane&15.
template <int K, int NC, bool RELU_A>
__global__ __launch_bounds__(256) void k_gemm(const float* __restrict__ A,
                                              const float* __restrict__ B,
                                              float* __restrict__ C, int M) {
  const int gw = (blockIdx.x * blockDim.x + threadIdx.x) >> 5;
  const int lane = threadIdx.x & 31;
  const int mt = (M + 15) >> 4;
  const int nt = NC / 16;
  const int tm = gw / nt;
  const int tn = gw % nt;
  if (tm >= mt) return;  // wave-uniform exit: EXEC stays all-ones for WMMA
  const int kh = lane >> 4;    // 0|1
  const int l15 = lane & 15;
  int arow = tm * 16 + l15;
  if (arow >= M) arow = M - 1;  // clamp loads in tail tile
  const int bcol = tn * 16 + l15;
  const float* ap = A + (size_t)arow * K;

  v8f acc = vzero8();
  for (int kk = 0; kk < K; kk += 4) {
    const int ka = kk + kh * 2;
    v2f a = *(const v2f*)(ap + ka);
    if (RELU_A) {
      a.x = fmaxf(a.x, 0.0f);
      a.y = fmaxf(a.y, 0.0f);
    }
    v2f b;
    b.x = B[(size_t)ka * NC + bcol];
    b.y = B[(size_t)(ka + 1) * NC + bcol];
    acc = wmma4(a, b, acc);
  }

  const int cr0 = tm * 16 + kh * 8;
#pragma unroll
  for (int r = 0; r < 8; ++r) {
    const int row = cr0 + r;
    if (row < M) C[(size_t)row * NC + tn * 16 + l15] = acc[r];
  }
}

// ---------------- self-loop init: out = xw * dinv^2 + bias ----------------
template <int NC>
__global__ void k_self_init(const float* __restrict__ xw,
                            const float* __restrict__ dinv,
                            const float* __restrict__ bias,
                            float* __restrict__ out, int n) {
  size_t i = (size_t)blockIdx.x * blockDim.x + threadIdx.x;
  if (i >= (size_t)n * NC) return;
  const int node = (int)(i / NC);
  const int j = (int)(i % NC);
  const float di = dinv[node];
  out[i] = xw[i] * di * di + bias[j];
}

// ---------------- edge scatter-add: out[dst] += xw[src] * dinv[s]*dinv[d] ------
// float4 gather (global_load_b128) + 4 hardware float atomics per thread.
// blockDim = 256: EPB edges per block, NC/4 threads per edge.
template <int NC, int EPB>
__global__ __launch_bounds__(256) void k_scatter(
    const int* __restrict__ src, const int* __restrict__ dst,
    const float* __restrict__ xw, const float* __restrict__ dinv,
    float* __restrict__ out, int E) {
  const int tpe = NC / 4;  // threads per edge
  const int le = threadIdx.x / tpe;
  const int j4 = (threadIdx.x % tpe) * 4;
  const long e = (long)blockIdx.x * EPB + le;
  if (e >= E) return;
  const int s = src[e];
  const int d = dst[e];
  const float norm = dinv[s] * dinv[d];
  const float4 v = *(const float4*)(xw + (size_t)s * NC + j4);
  float* o = out + (size_t)d * NC + j4;
  fadd_atomic(o + 0, v.x * norm);
  fadd_atomic(o + 1, v.y * norm);
  fadd_atomic(o + 2, v.z * norm);
  fadd_atomic(o + 3, v.w * norm);
}

// ---------------- fused edge MLP: relu(xe@We1+be1)@We2+be2 ----------------
// One wave handles 16 edge rows. GEMM1 (K=64 -> 256 cols) result goes through
// LDS (row-major, +4 pad => conflict-free column reads) so it can be re-read in
// the A-operand layout of GEMM2 (K=256 -> 64 cols). Avoids an 0.8 GB [E,256]
// intermediate in global memory.
#define EW 2  // waves per block; LDS = 2*16*260*4 = 33,280 B
__global__ __launch_bounds__(EW * 32) void k_edge_mlp(
    const float* __restrict__ xe, const float* __restrict__ We1,
    const float* __restrict__ be1, const float* __restrict__ We2,
    const float* __restrict__ be2, float* __restrict__ erec, int E) {
  __shared__ float inter[EW][16][260];
  const int wv = threadIdx.x >> 5;
  const int lane = threadIdx.x & 31;
  const int kh = lane >> 4;
  const int l15 = lane & 15;
  const long tile = (long)blockIdx.x * EW + wv;
  const long row0 = tile * 16;
  if (row0 >= E) return;  // wave-uniform
  long arow = row0 + l15;
  if (arow >= E) arow = E - 1;
  const float* ap = xe + (size_t)arow * 64;

  // GEMM1: inter[16,256] = relu(xe_tile @ We1 + be1)
  for (int tn = 0; tn < 16; ++tn) {
    v8f acc = vzero8();
    const int bcol = tn * 16 + l15;
    for (int kk = 0; kk < 64; kk += 4) {
      const int ka = kk + kh * 2;
      v2f a = *(const v2f*)(ap + ka);
      v2f b;
      b.x = We1[ka * 256 + bcol];
      b.y = We1[(ka + 1) * 256 + bcol];
      acc = wmma4(a, b, acc);
    }
    const float bv = be1[bcol];
#pragma unroll
    for (int r = 0; r < 8; ++r)
      inter[wv][kh * 8 + r][bcol] = fmaxf(acc[r] + bv, 0.0f);
  }

  // GEMM2: out[16,64] = inter @ We2 + be2  (A re-read from LDS in A-layout)
  for (int tn = 0; tn < 4; ++tn) {
    v8f acc = vzero8();
    const int bcol = tn * 16 + l15;
    for (int kk = 0; kk < 256; kk += 4) {
      const int ka = kk + kh * 2;
      v2f a = *(const v2f*)&inter[wv][l15][ka];
      v2f b;
      b.x = We2[ka * 64 + bcol];
      b.y = We2[(ka + 1) * 64 + bcol];
      acc = wmma4(a, b, acc);
    }
    const float bv = be2[bcol];
#pragma unroll
    for (int r = 0; r < 8; ++r) {
      const long row = row0 + kh * 8 + r;
      if (row < E) erec[(size_t)row * 64 + bcol] = acc[r] + bv;
    }
  }
}

extern "C" void kernel_launch(void* const* d_in, const int* in_sizes, int n_in,
                              void* d_out, int out_size, void* d_ws,
                              size_t ws_size, hipStream_t stream) {
  const float* x   = (const float*)d_in[0];   // [N,128]
  const int*   ei  = (const int*)d_in[1];     // [2,E] int32
  const float* xe  = (const float*)d_in[2];   // [E,64]
  const float* W1  = (const float*)d_in[3];   // [128,256]
  const float* b1  = (const float*)d_in[4];   // [256]
  const float* W2  = (const float*)d_in[5];   // [256,128]
  const float* b2  = (const float*)d_in[6];   // [128]
  const float* We1 = (const float*)d_in[7];   // [64,256]
  const float* be1 = (const float*)d_in[8];   // [256]
  const float* We2 = (const float*)d_in[9];   // [256,64]
  const float* be2 = (const float*)d_in[10];  // [64]

  const int N = in_sizes[0] / 128;
  const int E = in_sizes[1] / 2;
  const int* srcI = ei;
  const int* dstI = ei + E;

  float* x_rec = (float*)d_out;                  // [N,128]
  float* e_rec = x_rec + (size_t)N * 128;        // [E,64]

  // workspace: dinv[N] | bufA[N*256] (xw1, then hw2) | bufB[N*256] (h agg)
  float* dinv = (float*)d_ws;
  float* bufA = dinv + (((size_t)N + 63) & ~(size_t)63);
  float* bufB = bufA + (size_t)N * 256;

  // 1) degrees -> dinv
  k_deg_init<<<(N + 255) / 256, 256, 0, stream>>>(dinv, N);
  k_deg_edges<<<(E + 255) / 256, 256, 0, stream>>>(dstI, dinv, E);
  k_deg_rsqrt<<<(N + 255) / 256, 256, 0, stream>>>(dinv, N);

  // 2) xw1 = x @ W1          [N,256]
  {
    const int waves = ((N + 15) / 16) * (256 / 16);
    k_gemm<128, 256, false><<<(waves * 32 + 255) / 256, 256, 0, stream>>>(
        x, W1, bufA, N);
  }

  // 3) h_agg = self-loop + bias, then scatter edges   [N,256]
  {
    const size_t tot = (size_t)N * 256;
    k_self_init<256><<<(unsigned)((tot + 255) / 256), 256, 0, stream>>>(
        bufA, dinv, b1, bufB, N);
    k_scatter<256, 4><<<(E + 3) / 4, 256, 0, stream>>>(srcI, dstI, bufA, dinv,
                                                       bufB, E);
  }

  // 4) hw2 = relu(h_agg) @ W2     [N,128]  (ReLU fused into A-load)
  {
    const int waves = ((N + 15) / 16) * (128 / 16);
    k_gemm<256, 128, true><<<(waves * 32 + 255) / 256, 256, 0, stream>>>(
        bufB, W2, bufA, N);
  }

  // 5) x_rec = self-loop + bias, then scatter edges   [N,128]
  {
    const size_t tot = (size_t)N * 128;
    k_self_init<128><<<(unsigned)((tot + 255) / 256), 256, 0, stream>>>(
        bufA, dinv, b2, x_rec, N);
    k_scatter<128, 8><<<(E + 7) / 8, 256, 0, stream>>>(srcI, dstI, bufA, dinv,
                                                       x_rec, E);
  }

  // 6) fused edge MLP -> e_rec    [E,64]
  {
    const long tiles = ((long)E + 15) / 16;
    const unsigned blocks = (unsigned)((tiles + EW - 1) / EW);
    k_edge_mlp<<<blocks, EW * 32, 0, stream>>>(xe, We1, be1, We2, be2, e_rec, E);
  }
}